// Hierarchical_classifer_42331197669648
// MI455X (gfx1250) — compile-verified
//
#include <hip/hip_runtime.h>

typedef __attribute__((ext_vector_type(2))) float v2f;
typedef __attribute__((ext_vector_type(4))) float v4f;
typedef __attribute__((ext_vector_type(8))) float v8f;

#define DDIM 512
#define BATCH 16
#define N_NODES 1365
#define OUT_STRIDE (N_NODES * DDIM) /* 698880 floats per batch row */
#define LDS_STRIDE (DDIM + 4)       /* 516 floats: row stride 2064B -> bank-conflict-free b64 reads */

// One workgroup = 256 threads = 8 waves.
// Each wave computes TPW 16x16 output tiles (N columns) for one node.
// NSPLIT = 4/TPW workgroups cooperate on one node's N=512 columns.
template <int TPW>
__global__ __launch_bounds__(256) void tree_level_kernel(
    const float* __restrict__ W,   // [N_NODES, 512, 512]
    const float* __restrict__ bias,// [N_NODES, 512]
    const float* __restrict__ src, // inputs (level 0) or d_out (levels >= 1)
    long src_stride,               // floats between batch rows of src
    int parent_mode,               // 1: index src by parent node slot
    float* __restrict__ out,       // [16, N_NODES*512]
    int node_base,
    int apply_relu)
{
    constexpr int NSPLIT = 4 / TPW;
    __shared__ float lds_pa[BATCH * LDS_STRIDE];

    const int node  = node_base + (int)blockIdx.x / NSPLIT;
    const int slice = (int)blockIdx.x % NSPLIT;
    const int tid   = threadIdx.x;
    const int lane  = tid & 31;
    const int wave  = tid >> 5;
    const int m     = lane & 15;   // row within A-tile / col within B-tile
    const int h     = lane >> 4;   // lane half -> K sub-pair / M upper half

    // ---- stage parent activations (16 x 512 fp32 = 32KB) into LDS, coalesced b128 ----
    const float* pa = src;
    if (parent_mode) {
        const int parent = (node - 1) >> 2;
        pa = src + (long)parent * DDIM;
    }
#pragma unroll
    for (int i = 0; i < 8; ++i) {
        int idx = i * 256 + tid;            // 2048 float4 chunks total
        int row = idx >> 7;                 // 128 float4 per row
        int c4  = idx & 127;
        v4f v = *(const v4f*)(pa + (long)row * src_stride + c4 * 4);
        *(v4f*)&lds_pa[row * LDS_STRIDE + c4 * 4] = v;
    }
    __syncthreads();

    // ---- per-wave N-tiles ----
    const long wbase = (long)node * DDIM * DDIM;
    const int  n0    = slice * (128 * TPW) + wave * (16 * TPW);

    const float* Bp[TPW];
#pragma unroll
    for (int t = 0; t < TPW; ++t)
        Bp[t] = W + wbase + (long)(n0 + t * 16 + m) * DDIM + 2 * h;

    v8f acc[TPW];
#pragma unroll
    for (int t = 0; t < TPW; ++t) acc[t] = (v8f){};

    const float* aptr = &lds_pa[m * LDS_STRIDE + 2 * h];

    // K-loop: one fp32 WMMA per tile per 4 K-steps.
    // A frag: pa[m][k0+2h], pa[m][k0+2h+1]     (ds_load_b64)
    // B frag: W[node][n][k0+2h], [k0+2h+1]     (global_load_b64, row-contiguous stream)
#pragma unroll 4
    for (int k0 = 0; k0 < DDIM; k0 += 4) {
        v2f af = *(const v2f*)(aptr + k0);
#pragma unroll
        for (int t = 0; t < TPW; ++t) {
            v2f bf = *(const v2f*)(Bp[t] + k0);
            acc[t] = __builtin_amdgcn_wmma_f32_16x16x4_f32(
                /*neg_a=*/false, af, /*neg_b=*/false, bf,
                /*c_mod=*/(short)0, acc[t], /*reuse_a=*/false, /*reuse_b=*/false);
        }
    }

    // ---- epilogue: bias + relu + store (C/D layout: vgpr v, lane half h -> row v+8h, col n0+m) ----
    float* outb = out + (long)node * DDIM;
#pragma unroll
    for (int t = 0; t < TPW; ++t) {
        const int n = n0 + t * 16 + m;
        const float bv = bias[(long)node * DDIM + n];
#pragma unroll
        for (int v = 0; v < 8; ++v) {
            const int row = v + 8 * h;
            float val = acc[t][v] + bv;
            if (apply_relu) val = fmaxf(val, 0.0f);
            outb[(long)row * OUT_STRIDE + n] = val;
        }
    }
}

extern "C" void kernel_launch(void* const* d_in, const int* in_sizes, int n_in,
                              void* d_out, int out_size, void* d_ws, size_t ws_size,
                              hipStream_t stream) {
    const float* inputs = (const float*)d_in[0]; // [16, 512]
    const float* W      = (const float*)d_in[1]; // [1365, 512, 512]
    const float* b      = (const float*)d_in[2]; // [1365, 512]
    float*       out    = (float*)d_out;         // [16, 1365*512]

    // Level 0: 1 node, no relu, src = inputs (row stride 512). Split N 4-ways.
    tree_level_kernel<1><<<dim3(1 * 4), dim3(256), 0, stream>>>(
        W, b, inputs, (long)DDIM, /*parent_mode=*/0, out, /*node_base=*/0, /*relu=*/0);

    int base = 1, cnt = 4;
    for (int lvl = 1; lvl <= 5; ++lvl) {
        if (cnt < 256) {
            // few nodes: split each node's N=512 across 4 workgroups for occupancy
            tree_level_kernel<1><<<dim3(cnt * 4), dim3(256), 0, stream>>>(
                W, b, out, (long)OUT_STRIDE, 1, out, base, 1);
        } else {
            // plenty of nodes: one workgroup per node
            tree_level_kernel<4><<<dim3(cnt), dim3(256), 0, stream>>>(
                W, b, out, (long)OUT_STRIDE, 1, out, base, 1);
        }
        base += cnt;
        cnt *= 4;
    }
}